// GATv2Lightning_91104846283215
// MI455X (gfx1250) — compile-verified
//
#include <hip/hip_runtime.h>
#include <math.h>

typedef float v2f __attribute__((ext_vector_type(2)));
typedef float v8f __attribute__((ext_vector_type(8)));

#define HC 128      // heads * channels
#define NH 8        // heads
#define NCLS 10
#define SLOPE 0.2f

#define WMMA_F32(a, b, c) \
    __builtin_amdgcn_wmma_f32_16x16x4_f32(false, (a), false, (b), (short)0, (c), false, false)

// ---------------------------------------------------------------------------
// Fused dual GEMM: out0 = A @ W0, out1 = A @ W1  (A:[N,128], W:[128,128]).
// Block = 8 waves; each wave owns one 16-row tile and ALL 8 col tiles of both
// outputs (A fragment reused 16x per k-step). Both weight matrices are staged
// through LDS in two 64-row k-phases (64 KB static LDS total), so W is read
// from global exactly once per block and B fragments come from ds_load.
// ---------------------------------------------------------------------------
__global__ void __launch_bounds__(256)
gemm_dual_wmma(const float* __restrict__ A,
               const float* __restrict__ W0,
               const float* __restrict__ W1,
               float* __restrict__ out0,
               float* __restrict__ out1,
               int nTilesM) {
    __shared__ float ldsW0[64 * HC];   // 32 KB: 64 k-rows of W0
    __shared__ float ldsW1[64 * HC];   // 32 KB: 64 k-rows of W1

    const int tid    = threadIdx.x;
    const int waveId = tid >> 5;
    const int lane   = tid & 31;
    const int l16    = lane & 15;
    const int khalf  = lane >> 4;          // 0 or 1

    int tm = blockIdx.x * 8 + waveId;
    const bool active = tm < nTilesM;
    if (!active) tm = nTilesM - 1;         // clamp; stores masked below

    const float* Arow = A + (size_t)(tm * 16 + l16) * HC;  // A: M = lane%16
    __builtin_prefetch(Arow, 0, 0);                        // global_prefetch_b8

    v8f acc0[8] = {};
    v8f acc1[8] = {};

    for (int ph = 0; ph < 2; ++ph) {
        // ---- stage 64 k-rows of both weights into LDS (float4 moves) ----
        {
            const float4* w0 = (const float4*)(W0 + ph * 64 * HC);
            const float4* w1 = (const float4*)(W1 + ph * 64 * HC);
            float4* l0 = (float4*)ldsW0;
            float4* l1 = (float4*)ldsW1;
            for (int i = tid; i < 64 * HC / 4; i += 256) {
                l0[i] = w0[i];
                l1[i] = w1[i];
            }
        }
        __syncthreads();

        for (int k0 = 0; k0 < 64; k0 += 4) {
            const int k = k0 + 2 * khalf;                  // lane's K pair
            v2f a;                                         // contiguous b64
            a.x = Arow[ph * 64 + k];
            a.y = Arow[ph * 64 + k + 1];
#pragma unroll
            for (int tn = 0; tn < 8; ++tn) {
                const int col = tn * 16 + l16;
                v2f b0, b1;
                b0.x = ldsW0[k * HC + col];
                b0.y = ldsW0[(k + 1) * HC + col];
                b1.x = ldsW1[k * HC + col];
                b1.y = ldsW1[(k + 1) * HC + col];
                acc0[tn] = WMMA_F32(a, b0, acc0[tn]);
                acc1[tn] = WMMA_F32(a, b1, acc1[tn]);
            }
        }
        __syncthreads();    // all waves done with this phase before restage
    }

    if (!active) return;
    const int rbase = tm * 16 + 8 * khalf;                 // D: M = r + 8*(lane/16)
#pragma unroll
    for (int tn = 0; tn < 8; ++tn) {
        const int col = tn * 16 + l16;
#pragma unroll
        for (int r = 0; r < 8; ++r) {
            out0[(size_t)(rbase + r) * HC + col] = acc0[tn][r];
            out1[(size_t)(rbase + r) * HC + col] = acc1[tn][r];
        }
    }
}

// ---------------------------------------------------------------------------
// fc head: [N,128] x [128,10] + bias, WMMA with masked B/stores (cols 10..15).
// ---------------------------------------------------------------------------
__global__ void fc_wmma(const float* __restrict__ A,
                        const float* __restrict__ W,
                        const float* __restrict__ bias,
                        float* __restrict__ out,
                        int nTilesM) {
    const int wave = (blockIdx.x * blockDim.x + threadIdx.x) >> 5;
    const int lane = threadIdx.x & 31;
    if (wave >= nTilesM) return;
    const int l16   = lane & 15;
    const int khalf = lane >> 4;
    const int row = wave * 16 + l16;
    const int col = l16;

    v8f c = {};
#pragma unroll
    for (int k0 = 0; k0 < HC; k0 += 4) {
        const int k = k0 + 2 * khalf;
        v2f a;
        a.x = A[row * HC + k];
        a.y = A[row * HC + k + 1];
        v2f b;
        b.x = (col < NCLS) ? W[k * NCLS + col] : 0.0f;
        b.y = (col < NCLS) ? W[(k + 1) * NCLS + col] : 0.0f;
        c = WMMA_F32(a, b, c);
    }
    if (col < NCLS) {
        const int rbase = wave * 16 + 8 * khalf;
        const float bb = bias[col];
#pragma unroll
        for (int r = 0; r < 8; ++r)
            out[(rbase + r) * NCLS + col] = c[r] + bb;
    }
}

// ---------------------------------------------------------------------------
// Init: zero aggregation buffer, m = -inf, denom = 0.
// ---------------------------------------------------------------------------
__global__ void init_accum(float* __restrict__ agg, float* __restrict__ m,
                           float* __restrict__ denom, int N) {
    const int idx = blockIdx.x * blockDim.x + threadIdx.x;
    if (idx < N * HC) agg[idx] = 0.0f;
    if (idx < N * NH) { m[idx] = -INFINITY; denom[idx] = 0.0f; }
}

__device__ __forceinline__ void edge_ids(const int* __restrict__ ei, int e,
                                         int E, int& src, int& dst) {
    if (e < E) { src = ei[e]; dst = ei[E + e]; }
    else       { src = e - E; dst = e - E; }       // implicit self-loops
}

// per-lane partial of score: 4 channels per lane, 4 lanes per head
__device__ __forceinline__ float edge_partial(const float4 a4, const float4 b4,
                                              const float4 w4) {
    float s = 0.0f, t;
    t = a4.x + b4.x; s += w4.x * (t > 0.0f ? t : SLOPE * t);
    t = a4.y + b4.y; s += w4.y * (t > 0.0f ? t : SLOPE * t);
    t = a4.z + b4.z; s += w4.z * (t > 0.0f ? t : SLOPE * t);
    t = a4.w + b4.w; s += w4.w * (t > 0.0f ? t : SLOPE * t);
    return s;
}

// ---------------------------------------------------------------------------
// Pass 1: per-edge, per-head score; atomicMax into m[dst,h]. One wave per edge.
// ---------------------------------------------------------------------------
__global__ void edge_score_max(const float* __restrict__ xl,
                               const float* __restrict__ xr,
                               const int* __restrict__ ei,
                               const float* __restrict__ att,
                               float* __restrict__ m, int E, int N) {
    const int wave = (blockIdx.x * blockDim.x + threadIdx.x) >> 5;
    const int lane = threadIdx.x & 31;
    if (wave >= E + N) return;
    int src, dst;
    edge_ids(ei, wave, E, src, dst);

    const float4 a4 = *(const float4*)(xl + src * HC + 4 * lane);
    const float4 b4 = *(const float4*)(xr + dst * HC + 4 * lane);
    const float4 w4 = *(const float4*)(att + 4 * lane);   // att is [8,16] flat
    float s = edge_partial(a4, b4, w4);
    s += __shfl_xor(s, 1, 32);                            // reduce within head
    s += __shfl_xor(s, 2, 32);
    if ((lane & 3) == 0)
        atomicMax(m + dst * NH + (lane >> 2), s);
}

// ---------------------------------------------------------------------------
// Pass 2: p = exp(score - m[dst]); accumulate p*xl[src] and denom via atomics.
// ---------------------------------------------------------------------------
__global__ void edge_accumulate(const float* __restrict__ xl,
                                const float* __restrict__ xr,
                                const int* __restrict__ ei,
                                const float* __restrict__ att,
                                const float* __restrict__ m,
                                float* __restrict__ agg,
                                float* __restrict__ denom, int E, int N) {
    const int wave = (blockIdx.x * blockDim.x + threadIdx.x) >> 5;
    const int lane = threadIdx.x & 31;
    if (wave >= E + N) return;
    int src, dst;
    edge_ids(ei, wave, E, src, dst);
    const int h = lane >> 2;

    const float4 a4 = *(const float4*)(xl + src * HC + 4 * lane);
    const float4 b4 = *(const float4*)(xr + dst * HC + 4 * lane);
    const float4 w4 = *(const float4*)(att + 4 * lane);
    float s = edge_partial(a4, b4, w4);
    s += __shfl_xor(s, 1, 32);
    s += __shfl_xor(s, 2, 32);                 // all 4 lanes of a head hold s

    const float p = __expf(s - m[dst * NH + h]);
    float* o = agg + dst * HC + 4 * lane;
    atomicAdd(o + 0, p * a4.x);
    atomicAdd(o + 1, p * a4.y);
    atomicAdd(o + 2, p * a4.z);
    atomicAdd(o + 3, p * a4.w);
    if ((lane & 3) == 0)
        atomicAdd(denom + dst * NH + h, p);
}

// ---------------------------------------------------------------------------
// Pass 3: normalize by denom, add bias, optional ReLU (in place).
// ---------------------------------------------------------------------------
__global__ void normalize_bias_act(float* __restrict__ agg,
                                   const float* __restrict__ denom,
                                   const float* __restrict__ bias,
                                   int N, int doRelu) {
    const int idx = blockIdx.x * blockDim.x + threadIdx.x;
    if (idx >= N * HC) return;
    const int node = idx >> 7;
    const int col  = idx & (HC - 1);
    const int h    = col >> 4;
    float v = agg[idx] / (denom[node * NH + h] + 1e-16f) + bias[col];
    if (doRelu) v = fmaxf(v, 0.0f);
    agg[idx] = v;
}

// ---------------------------------------------------------------------------
extern "C" void kernel_launch(void* const* d_in, const int* in_sizes, int n_in,
                              void* d_out, int out_size, void* d_ws, size_t ws_size,
                              hipStream_t stream) {
    const float* x    = (const float*)d_in[0];
    const int*   ei   = (const int*)  d_in[1];
    const float* Wl1  = (const float*)d_in[2];
    const float* Wr1  = (const float*)d_in[3];
    const float* att1 = (const float*)d_in[4];
    const float* b1   = (const float*)d_in[5];
    const float* Wl2  = (const float*)d_in[6];
    const float* Wr2  = (const float*)d_in[7];
    const float* att2 = (const float*)d_in[8];
    const float* b2   = (const float*)d_in[9];
    const float* Wfc  = (const float*)d_in[10];
    const float* bfc  = (const float*)d_in[11];
    float* out = (float*)d_out;

    const int N = in_sizes[0] / HC;     // 50000
    const int E = in_sizes[1] / 2;      // 800000
    const int Etot = E + N;             // + self loops

    float* xl    = (float*)d_ws;
    float* xr    = xl  + (size_t)N * HC;
    float* agg   = xr  + (size_t)N * HC;
    float* m     = agg + (size_t)N * HC;
    float* denom = m   + (size_t)N * NH;

    const int tilesM   = N / 16;                        // N divisible by 16
    const dim3 blk(256);
    const int dualGrid = (tilesM + 7) / 8;              // 8 row tiles per block
    const int fcGrid   = (tilesM + 7) / 8;
    const int edgeGrid = (Etot + 7) / 8;                // one wave per edge
    const int nodeGrid = (N * HC + 255) / 256;

    // ---- layer 1 ----
    gemm_dual_wmma<<<dualGrid, blk, 0, stream>>>(x, Wl1, Wr1, xl, xr, tilesM);
    init_accum<<<nodeGrid, blk, 0, stream>>>(agg, m, denom, N);
    edge_score_max<<<edgeGrid, blk, 0, stream>>>(xl, xr, ei, att1, m, E, N);
    edge_accumulate<<<edgeGrid, blk, 0, stream>>>(xl, xr, ei, att1, m, agg, denom, E, N);
    normalize_bias_act<<<nodeGrid, blk, 0, stream>>>(agg, denom, b1, N, 1);

    // ---- layer 2 (input = agg, i.e. relu(layer1)) ----
    gemm_dual_wmma<<<dualGrid, blk, 0, stream>>>(agg, Wl2, Wr2, xl, xr, tilesM);
    init_accum<<<nodeGrid, blk, 0, stream>>>(agg, m, denom, N);
    edge_score_max<<<edgeGrid, blk, 0, stream>>>(xl, xr, ei, att2, m, E, N);
    edge_accumulate<<<edgeGrid, blk, 0, stream>>>(xl, xr, ei, att2, m, agg, denom, E, N);
    normalize_bias_act<<<nodeGrid, blk, 0, stream>>>(agg, denom, b2, N, 1);

    // ---- fc head ----
    fc_wmma<<<fcGrid, blk, 0, stream>>>(agg, Wfc, bfc, out, tilesM);
}